// SequentialFocalLoss_61813169324397
// MI455X (gfx1250) — compile-verified
//
#include <hip/hip_runtime.h>
#include <hip/hip_bf16.h>

typedef __attribute__((ext_vector_type(2))) float v2f;
typedef __attribute__((ext_vector_type(8))) float v8f;

#define BB   512
#define SS   8192
#define CC   5
#define BLK  256
#define CHUNKS (SS / BLK)            // 32
#define NBLOCKS (BB * CHUNKS)        // 16384
#define TW   0.2f                    // TRANSITION_WEIGHT

// Transition matrix row r, column c (padded to 16 columns with zeros).
__device__ __forceinline__ float tval(int r, int c) {
    switch (r) {
        case 0:  return c == 0 ? 0.8f : c == 1 ? 0.2f : 0.0f;
        case 1:  return c == 0 ? 0.1f : c == 1 ? 0.4f : c == 2 ? 0.4f : c == 4 ? 0.1f : 0.0f;
        case 2:  return c == 1 ? 0.1f : c == 2 ? 0.7f : c == 3 ? 0.2f : 0.0f;
        case 3:  return c == 2 ? 0.3f : c == 3 ? 0.7f : 0.0f;
        default: return c == 0 ? 0.1f : c == 1 ? 0.2f : c == 4 ? 0.7f : 0.0f; // row 4
    }
}

__global__ __launch_bounds__(BLK)
void focal_main(const float* __restrict__ inputs, const int* __restrict__ targets,
                float* __restrict__ part) {
    __shared__ __align__(16) float s_logit[(BLK + 1) * CC + 3];  // 1285 (+pad to 16B)
    // s_lp padded: KL phase reads up to index 256*5 + 15 = 1295; pad region holds zeros.
    __shared__ float s_lp[(BLK + 1) * CC + 27];                  // 1312
    __shared__ float s_red[2 * BLK];

    const int tid   = threadIdx.x;
    const int bidx  = blockIdx.x;
    const int b     = bidx >> 5;          // row
    const int t0    = (bidx & 31) << 8;   // chunk start timestep
    const size_t baseF  = ((size_t)b * SS + t0) * CC; // float index of tile start (16B aligned)
    const size_t totalF = (size_t)BB * SS * CC;

    // ---- Stage 257 timesteps x 5 logits into LDS via async copy (memory -> LDS, no VGPRs) ----
    float4* sl4 = reinterpret_cast<float4*>(s_logit);
    const uint32_t ldsBase = (uint32_t)(uintptr_t)s_logit;       // low 32 bits = LDS offset
    const uint64_t gBase   = (uint64_t)(uintptr_t)(inputs + baseF);
    for (int i = tid; i < 321; i += BLK) {                       // 321*16B = 1284 floats
        if (baseF + (size_t)i * 4 + 4 <= totalF) {
            uint64_t ga = gBase + (uint64_t)i * 16u;
            uint32_t la = ldsBase + (uint32_t)i * 16u;
            asm volatile("global_load_async_to_lds_b128 %0, %1, off"
                         :: "v"(la), "v"(ga) : "memory");
        } else {                                                 // only final block's tail
            float4 z; z.x = 0.f; z.y = 0.f; z.z = 0.f; z.w = 0.f;
            sl4[i] = z;
        }
    }
    if (tid == 0) {
        size_t g = baseF + 1284;
        s_logit[1284] = (g < totalF) ? inputs[g] : 0.0f;
    }
    if (tid < 27) s_lp[(BLK + 1) * CC + tid] = 0.0f;             // zero the pad region
    asm volatile("s_wait_asynccnt 0x0" ::: "memory");
    __syncthreads();

    // ---- Per-thread log-softmax over 5 classes ----
    float x[CC], lp[CC], p[CC];
#pragma unroll
    for (int c = 0; c < CC; ++c) x[c] = s_logit[tid * CC + c];
    float mx = x[0];
#pragma unroll
    for (int c = 1; c < CC; ++c) mx = fmaxf(mx, x[c]);
    float sum = 0.0f, e[CC];
#pragma unroll
    for (int c = 0; c < CC; ++c) { e[c] = __expf(x[c] - mx); sum += e[c]; }
    float inv = 1.0f / sum;
    float ls  = __logf(sum);
#pragma unroll
    for (int c = 0; c < CC; ++c) { lp[c] = x[c] - mx - ls; p[c] = e[c] * inv; }
#pragma unroll
    for (int c = 0; c < CC; ++c) s_lp[tid * CC + c] = lp[c];

    // Thread 0 also produces log-probs for the boundary timestep (index BLK).
    if (tid == 0) {
        float y[CC];
#pragma unroll
        for (int c = 0; c < CC; ++c) y[c] = s_logit[BLK * CC + c];
        float m2 = y[0];
#pragma unroll
        for (int c = 1; c < CC; ++c) m2 = fmaxf(m2, y[c]);
        float s2 = 0.0f;
#pragma unroll
        for (int c = 0; c < CC; ++c) s2 += __expf(y[c] - m2);
        float l2 = __logf(s2);
#pragma unroll
        for (int c = 0; c < CC; ++c) s_lp[BLK * CC + c] = y[c] - m2 - l2;
    }

    // ---- Focal loss term (every timestep contributes) ----
    int   tg = targets[(size_t)b * SS + t0 + tid];
    float tl = (tg == 0) ? lp[0] : (tg == 1) ? lp[1] : (tg == 2) ? lp[2]
             : (tg == 3) ? lp[3] : lp[4];
    float pt = __expf(tl);
    float om = 1.0f - pt;
    float focal = om * om * (-tl);            // GAMMA = 2

    __syncthreads();                          // s_lp fully populated

    // ---- expected = probs @ T via V_WMMA_F32_16X16X4_F32 ----
    // Each wave handles 32 timesteps = two 16-row A-tiles; K=5 split as K=4 + rank-1 (class 4).
    const int lane    = tid & 31;
    const int w       = tid >> 5;
    const int laneLow = lane & 15;
    const bool hi     = lane >= 16;

    // Tile 0 (rows = timesteps of lanes 0..15). A layout: V0 = {K0 | K2}, V1 = {K1 | K3}.
    float sh_p2 = __shfl(p[2], laneLow, 32);
    float sh_p3 = __shfl(p[3], laneLow, 32);
    v2f aA; aA[0] = hi ? sh_p2 : p[0];  aA[1] = hi ? sh_p3 : p[1];   // classes 0..3
    v2f aB; aB[0] = hi ? 0.0f  : p[4];  aB[1] = 0.0f;                // class 4 at K=0, rest 0

    // Tile 1 (rows = timesteps of lanes 16..31).
    float t1p0 = __shfl(p[0], lane | 16, 32);
    float t1p1 = __shfl(p[1], lane | 16, 32);
    float t1p4 = __shfl(p[4], lane | 16, 32);
    v2f cA; cA[0] = hi ? p[2] : t1p0;   cA[1] = hi ? p[3] : t1p1;
    v2f cB; cB[0] = hi ? 0.0f : t1p4;   cB[1] = 0.0f;

    // B = T padded to 4x16 (rows 0..3), second chunk row0 = T row 4.
    v2f bA; bA[0] = hi ? tval(2, laneLow) : tval(0, laneLow);
            bA[1] = hi ? tval(3, laneLow) : tval(1, laneLow);
    v2f bB; bB[0] = hi ? 0.0f : tval(4, laneLow);  bB[1] = 0.0f;

    v8f acc0 = {};
    acc0 = __builtin_amdgcn_wmma_f32_16x16x4_f32(false, aA, false, bA, (short)0, acc0, false, false);
    acc0 = __builtin_amdgcn_wmma_f32_16x16x4_f32(false, aB, false, bB, (short)0, acc0, false, false);
    v8f acc1 = {};
    acc1 = __builtin_amdgcn_wmma_f32_16x16x4_f32(false, cA, false, bA, (short)0, acc1, false, false);
    acc1 = __builtin_amdgcn_wmma_f32_16x16x4_f32(false, cB, false, bB, (short)0, acc1, false, false);

    // ---- KL: sum expected*(log expected - logp_next) ----
    // Branch-free: for padded columns (n>=5) ev==0 exactly, so the term self-cancels
    // (pad region of s_lp holds finite zeros; eps inside log avoids 0*inf).
    float kl = 0.0f;
    const int c    = laneLow;
    const int mofs = hi ? 8 : 0;
#pragma unroll
    for (int r = 0; r < 8; ++r) {
        {   // tile 0: local timestep lt = w*32 + (r + mofs)
            int   lt  = w * 32 + r + mofs;
            float ev  = acc0[r];
            float lg  = __logf(fmaxf(ev, 1e-30f));
            float lpn = s_lp[(lt + 1) * CC + c];
            float ct  = ev * (lg - lpn);
            kl += (t0 + lt < SS - 1) ? ct : 0.0f;
        }
        {   // tile 1: lt = w*32 + 16 + (r + mofs)
            int   lt  = w * 32 + 16 + r + mofs;
            float ev  = acc1[r];
            float lg  = __logf(fmaxf(ev, 1e-30f));
            float lpn = s_lp[(lt + 1) * CC + c];
            float ct  = ev * (lg - lpn);
            kl += (t0 + lt < SS - 1) ? ct : 0.0f;
        }
    }

    // ---- Deterministic block reduction ----
    s_red[tid]        = focal;
    s_red[BLK + tid]  = kl;
    __syncthreads();
    for (int s2 = BLK / 2; s2 > 0; s2 >>= 1) {
        if (tid < s2) {
            s_red[tid]       += s_red[tid + s2];
            s_red[BLK + tid] += s_red[BLK + tid + s2];
        }
        __syncthreads();
    }
    if (tid == 0) {
        part[2 * bidx]     = s_red[0];
        part[2 * bidx + 1] = s_red[BLK];
    }
}

__global__ __launch_bounds__(BLK)
void focal_combine(const float* __restrict__ part, float* __restrict__ out) {
    __shared__ float s_red[2 * BLK];
    const int tid = threadIdx.x;
    float fa = 0.0f, ka = 0.0f;
    for (int i = tid; i < NBLOCKS; i += BLK) {
        fa += part[2 * i];
        ka += part[2 * i + 1];
    }
    s_red[tid]       = fa;
    s_red[BLK + tid] = ka;
    __syncthreads();
    for (int s2 = BLK / 2; s2 > 0; s2 >>= 1) {
        if (tid < s2) {
            s_red[tid]       += s_red[tid + s2];
            s_red[BLK + tid] += s_red[BLK + tid + s2];
        }
        __syncthreads();
    }
    if (tid == 0) {
        float class_loss      = s_red[0]   / (float)((size_t)BB * SS);
        float transition_loss = s_red[BLK] / (float)((size_t)BB * (SS - 1));
        out[0] = class_loss + TW * transition_loss;
    }
}

extern "C" void kernel_launch(void* const* d_in, const int* in_sizes, int n_in,
                              void* d_out, int out_size, void* d_ws, size_t ws_size,
                              hipStream_t stream) {
    (void)in_sizes; (void)n_in; (void)out_size; (void)ws_size;
    const float* inputs  = (const float*)d_in[0];
    const int*   targets = (const int*)d_in[1];
    float*       part    = (float*)d_ws;    // NBLOCKS * 2 floats = 128 KB
    float*       out     = (float*)d_out;

    focal_main<<<dim3(NBLOCKS), dim3(BLK), 0, stream>>>(inputs, targets, part);
    focal_combine<<<dim3(1), dim3(BLK), 0, stream>>>(part, out);
}